// GCNInference_23089744183867
// MI455X (gfx1250) — compile-verified
//
#include <hip/hip_runtime.h>

typedef __attribute__((ext_vector_type(2))) float v2f;
typedef __attribute__((ext_vector_type(8))) float v8f;

static constexpr int kNodes = 50000;
static constexpr int kEdges = 800000;
static constexpr int kDHid  = 128;   // D_IN == D_HID == 128
static constexpr int kDOut  = 64;
static constexpr int kMT    = 5;     // M-tiles per wave: 50000 / (5*16) = 625 exactly

// ---------------------------------------------------------------------------
// Zero-fill (float4 stores)
// ---------------------------------------------------------------------------
__global__ __launch_bounds__(256) void zero_f4(float4* __restrict__ p, int n4) {
  int i = blockIdx.x * blockDim.x + threadIdx.x;
  if (i < n4) p[i] = make_float4(0.f, 0.f, 0.f, 0.f);
}

// ---------------------------------------------------------------------------
// SpMM scatter: one wave (32 lanes) per edge, 4 features per lane.
//   out[dst, :] += w * feat[src, :]
// Gathers hit L2 (feature matrix is 25.6MB < 192MB L2); scatter uses native
// f32 global atomics (global_atomic_add_f32, no-return -> STOREcnt path).
// Wave count == kEdges exactly (no divergence / EXEC holes).
// ---------------------------------------------------------------------------
__global__ __launch_bounds__(256) void spmm_scatter(
    const int* __restrict__ edge_index,     // [2, E] : row0 = dst, row1 = src
    const float* __restrict__ edge_weight,  // [E]
    const float* __restrict__ feat,         // [N, 128]
    float* __restrict__ out) {              // [N, 128] pre-zeroed
  const int wave = (blockIdx.x * blockDim.x + threadIdx.x) >> 5;
  const int lane = threadIdx.x & 31;
  const int dst = edge_index[wave];
  const int src = edge_index[kEdges + wave];
  const float w = edge_weight[wave];
  const float4 f = ((const float4*)(feat + (size_t)src * kDHid))[lane];
  float* o = out + (size_t)dst * kDHid + lane * 4;
  unsafeAtomicAdd(o + 0, f.x * w);
  unsafeAtomicAdd(o + 1, f.y * w);
  unsafeAtomicAdd(o + 2, f.z * w);
  unsafeAtomicAdd(o + 3, f.w * w);
}

// ---------------------------------------------------------------------------
// C[M,N] = relu?(A[M,128] @ W[128,N] + bias[N]) via V_WMMA_F32_16X16X4_F32.
//
// Each wave computes kMT (=5) vertically-stacked 16x16 tiles in one N-column
// band: the B fragment loaded per K-step is reused by 5 independent WMMAs
// (amortizes the strided B loads 5x and keeps 5 independent accumulator
// chains in flight to cover XDL latency). threadIdx.y selects the N-tile.
//
// Per-lane operand layout (ISA 7.12.2, 32-bit matrices, wave32):
//   A (16x4): lanes 0-15 row=lane, K={0,1}; lanes 16-31 row=lane-16, K={2,3}
//             -> one aligned float2 load per (tile, K-step).
//   B (4x16): lanes 0-15 col=lane, K={0,1}; lanes 16-31 K={2,3}.
//   C/D (16x16, 8 VGPRs): VGPR i -> row = i + 8*(lane>>4), col = lane&15.
// ---------------------------------------------------------------------------
template <int N, bool RELU>
__global__ void gemm_bias_act(const float* __restrict__ A,
                              const float* __restrict__ W,     // [128, N]
                              const float* __restrict__ bias,  // [N]
                              float* __restrict__ C) {         // [M, N]
  const int lane = threadIdx.x;        // 0..31
  const int col  = lane & 15;
  const int hi   = lane >> 4;          // 0 or 1
  const int rowBase = blockIdx.x * (16 * kMT);
  const int colBase = threadIdx.y * 16;

  const float* Arow = A + (size_t)(rowBase + col) * kDHid;  // + m*16*kDHid via imm offset
  const float* Wc   = W + colBase + col;

  v8f acc[kMT];
#pragma unroll
  for (int m = 0; m < kMT; ++m) acc[m] = (v8f){};

#pragma unroll
  for (int k = 0; k < kDHid; k += 4) {
    const int kk = k + 2 * hi;
    v2f b;
    b.x = Wc[(size_t)kk * N];
    b.y = Wc[(size_t)(kk + 1) * N];
#pragma unroll
    for (int m = 0; m < kMT; ++m) {
      v2f a = *(const v2f*)(Arow + (size_t)m * 16 * kDHid + kk);
      // 8 args: (neg_a, A, neg_b, B, c_mod, C, reuse_a, reuse_b)
      acc[m] = __builtin_amdgcn_wmma_f32_16x16x4_f32(
          false, a, false, b, (short)0, acc[m], false, false);
    }
  }

  const float bb = bias[colBase + col];
  float* Cc = C + colBase + col;
#pragma unroll
  for (int m = 0; m < kMT; ++m) {
#pragma unroll
    for (int i = 0; i < 8; ++i) {
      float v = acc[m][i] + bb;
      if (RELU) v = fmaxf(v, 0.f);
      Cc[(size_t)(rowBase + m * 16 + hi * 8 + i) * N] = v;
    }
  }
}

// ---------------------------------------------------------------------------
// Driver: 3 GCN layers.  ws layout: Agg = aggregation buffer, H = hidden
// activations (2 x 50000 x 128 f32 = 51.2 MB total).
// ---------------------------------------------------------------------------
extern "C" void kernel_launch(void* const* d_in, const int* in_sizes, int n_in,
                              void* d_out, int out_size, void* d_ws, size_t ws_size,
                              hipStream_t stream) {
  const float* x  = (const float*)d_in[0];
  const int*   ei = (const int*)d_in[1];
  const float* ew = (const float*)d_in[2];
  const float* W1 = (const float*)d_in[3];
  const float* b1 = (const float*)d_in[4];
  const float* W2 = (const float*)d_in[5];
  const float* b2 = (const float*)d_in[6];
  const float* W3 = (const float*)d_in[7];
  const float* b3 = (const float*)d_in[8];
  float* out = (float*)d_out;

  float* Agg = (float*)d_ws;
  float* H   = Agg + (size_t)kNodes * kDHid;

  const int n4 = kNodes * kDHid / 4;
  const dim3 zgrid((n4 + 255) / 256), zblk(256);
  const dim3 sgrid(kEdges / 8), sblk(256);          // 8 waves/block, 1 edge/wave
  const dim3 ggrid(kNodes / (16 * kMT));            // 625 blocks exactly

  // Layer 1: Agg = adj @ x ; H = relu(Agg @ W1 + b1)
  zero_f4<<<zgrid, zblk, 0, stream>>>((float4*)Agg, n4);
  spmm_scatter<<<sgrid, sblk, 0, stream>>>(ei, ew, x, Agg);
  gemm_bias_act<kDHid, true><<<ggrid, dim3(32, kDHid / 16), 0, stream>>>(Agg, W1, b1, H);

  // Layer 2
  zero_f4<<<zgrid, zblk, 0, stream>>>((float4*)Agg, n4);
  spmm_scatter<<<sgrid, sblk, 0, stream>>>(ei, ew, H, Agg);
  gemm_bias_act<kDHid, true><<<ggrid, dim3(32, kDHid / 16), 0, stream>>>(Agg, W2, b2, H);

  // Layer 3 (no ReLU, N=64) -> d_out
  zero_f4<<<zgrid, zblk, 0, stream>>>((float4*)Agg, n4);
  spmm_scatter<<<sgrid, sblk, 0, stream>>>(ei, ew, H, Agg);
  gemm_bias_act<kDOut, false><<<ggrid, dim3(32, kDOut / 16), 0, stream>>>(Agg, W3, b3, out);
}